// LightGCN_39238821216967
// MI455X (gfx1250) — compile-verified
//
#include <hip/hip_runtime.h>

#define THREADS 256
static const int LAYERS = 100;

#ifndef __has_builtin
#define __has_builtin(x) 0
#endif

#define AS1 __attribute__((address_space(1)))
#define AS3 __attribute__((address_space(3)))

// gfx1250 async-to-LDS builtin takes typed vector pointers: 2 x i32 for b64.
typedef int v2i __attribute__((vector_size(2 * sizeof(int))));

#if defined(__AMDGCN__) && __has_builtin(__builtin_amdgcn_global_load_async_to_lds_b64)
#define USE_ASYNC_LDS 1
#else
#define USE_ASYNC_LDS 0
#endif

#if __has_builtin(__builtin_amdgcn_s_wait_asynccnt)
#define ASYNC_WAIT(n)                                   \
  do {                                                  \
    __builtin_amdgcn_s_wait_asynccnt(n);                \
    asm volatile("" ::: "memory");                      \
  } while (0)
#else
#define ASYNC_WAIT(n) asm volatile("s_wait_asynccnt %0" ::"i"(n) : "memory")
#endif

// ---------------- precompute kernels ----------------

__global__ void k_zero_i32(int* __restrict__ p, int n) {
  int i = blockIdx.x * blockDim.x + threadIdx.x;
  if (i < n) p[i] = 0;
}

__global__ void k_count_deg(const int* __restrict__ col, int* __restrict__ deg, int e) {
  int i = blockIdx.x * blockDim.x + threadIdx.x;
  if (i < e) atomicAdd(&deg[col[i]], 1);
}

__global__ void k_dinv(const int* __restrict__ deg, float* __restrict__ dinv, int n) {
  int i = blockIdx.x * blockDim.x + threadIdx.x;
  if (i < n) {
    int d = deg[i];
    dinv[i] = (d > 0) ? rsqrtf((float)d) : 0.0f;
  }
}

// Exclusive scan of deg, chunked: each block scans a 1024-element chunk.
__global__ void k_scan_block(const int* __restrict__ deg, int* __restrict__ chunkExcl,
                             int* __restrict__ part, int n) {
  __shared__ int sh[THREADS];
  int t = threadIdx.x;
  int base = blockIdx.x * (THREADS * 4);
  int v[4], pre[4];
  int s = 0;
#pragma unroll
  for (int k = 0; k < 4; ++k) {
    int idx = base + t * 4 + k;
    v[k] = (idx < n) ? deg[idx] : 0;
    pre[k] = s;
    s += v[k];
  }
  sh[t] = s;
  __syncthreads();
  for (int o = 1; o < THREADS; o <<= 1) {
    int x = (t >= o) ? sh[t - o] : 0;
    __syncthreads();
    sh[t] += x;
    __syncthreads();
  }
  int excl = sh[t] - s;  // exclusive across threads of this chunk
#pragma unroll
  for (int k = 0; k < 4; ++k) {
    int idx = base + t * 4 + k;
    if (idx < n) chunkExcl[idx] = excl + pre[k];
  }
  if (t == THREADS - 1) part[blockIdx.x] = sh[t];
}

// Single-block exclusive scan of per-chunk totals (m <= 256 holds for N <= 262144).
__global__ void k_scan_part(const int* __restrict__ part, int* __restrict__ partExcl, int m) {
  __shared__ int sh[THREADS];
  int t = threadIdx.x;
  int v = (t < m) ? part[t] : 0;
  sh[t] = v;
  __syncthreads();
  for (int o = 1; o < THREADS; o <<= 1) {
    int x = (t >= o) ? sh[t - o] : 0;
    __syncthreads();
    sh[t] += x;
    __syncthreads();
  }
  if (t < m) partExcl[t] = sh[t] - v;
}

__global__ void k_scan_add(const int* __restrict__ chunkExcl, const int* __restrict__ partExcl,
                           int* __restrict__ off, int* __restrict__ cursor, int n, int e) {
  int i = blockIdx.x * blockDim.x + threadIdx.x;
  if (i < n) {
    int o = chunkExcl[i] + partExcl[i >> 10];
    off[i] = o;
    cursor[i] = o;
  }
  if (i == 0) off[n] = e;
}

// Scatter edges into CSC slots; pair = {source row, precomputed norm}.
__global__ void k_fill(const int* __restrict__ row, const int* __restrict__ col,
                       const float* __restrict__ dinv, int* __restrict__ cursor,
                       int2* __restrict__ pairs, int e) {
  int i = blockIdx.x * blockDim.x + threadIdx.x;
  if (i < e) {
    int r = row[i], c = col[i];
    int p = atomicAdd(&cursor[c], 1);
    float w = dinv[r] * dinv[c];
    pairs[p] = make_int2(r, __float_as_int(w));
  }
}

__global__ void k_init(const float* __restrict__ emb, float* __restrict__ x0,
                       float* __restrict__ acc, int n) {
  int i = blockIdx.x * blockDim.x + threadIdx.x;
  if (i < n) {
    float v = emb[i];
    x0[i] = v;
    acc[i] = v;
  }
}

// ---------------- main layer kernel ----------------
// One wave32 per destination node; lane l owns dims [2l, 2l+1] as a float2.
// Edge {row,norm} pairs are staged into per-wave LDS chunk buffers — via
// GLOBAL_LOAD_ASYNC_TO_LDS_B64 (double-buffered, ASYNCcnt-ordered) when the
// toolchain exposes the builtin, else via coalesced load + ds_store — and
// consumed with broadcast ds_load_b64. x_new accumulates in registers
// (no global atomics); acc += is fused in.
__device__ __forceinline__ void stage_pair_async(const int2* gsrc, int2* ldst) {
#if USE_ASYNC_LDS
  __builtin_amdgcn_global_load_async_to_lds_b64(
      (AS1 v2i*)(unsigned long long)gsrc,  // global source (2 x i32, addrspace 1)
      (AS3 v2i*)ldst,                      // LDS destination (2 x i32, addrspace 3)
      0, 0);
#else
  (void)gsrc;
  (void)ldst;
#endif
}

__global__ __launch_bounds__(THREADS) void k_layer(
    const float2* __restrict__ xin, float2* __restrict__ xout, float2* __restrict__ acc,
    const int2* __restrict__ pairs, const int* __restrict__ off, int n) {
  __shared__ int2 ebuf[THREADS / 32][2][32];
  int wslot = threadIdx.x >> 5;
  int lane  = threadIdx.x & 31;
  int node  = (int)((blockIdx.x * (unsigned)THREADS + threadIdx.x) >> 5);
  if (node >= n) return;

  int s = off[node];
  int e = off[node + 1];
  float2 sum = make_float2(0.0f, 0.0f);

#if USE_ASYNC_LDS
  // Prologue: stage first chunk asynchronously.
  if (s < e) {
    int idx = s + lane;
    if (idx < e) stage_pair_async(&pairs[idx], &ebuf[wslot][0][lane]);
  }
  int cur = 0;
  for (int base = s; base < e; base += 32) {
    int nxt = base + 32;
    if (nxt < e) {
      int idx = nxt + lane;
      if (idx < e) stage_pair_async(&pairs[idx], &ebuf[wslot][cur ^ 1][lane]);
      ASYNC_WAIT(1);  // current chunk landed (in-order completion), next in flight
    } else {
      ASYNC_WAIT(0);
    }
    int cnt = min(32, e - base);
    const int2* buf = &ebuf[wslot][cur][0];
#pragma unroll 4
    for (int j = 0; j < cnt; ++j) {
      int2 q = buf[j];                 // broadcast ds_load_b64
      float w = __int_as_float(q.y);
      float2 xv = xin[(size_t)q.x * 32 + lane];
      sum.x = fmaf(w, xv.x, sum.x);
      sum.y = fmaf(w, xv.y, sum.y);
    }
    cur ^= 1;
    asm volatile("" ::: "memory");
  }
#else
  for (int base = s; base < e; base += 32) {
    int idx = base + lane;
    if (idx < e) ebuf[wslot][0][lane] = pairs[idx];  // coalesced load + ds_store
    if (base + 32 < e) __builtin_prefetch(&pairs[base + 32 + lane], 0, 3);
    __builtin_amdgcn_wave_barrier();
    asm volatile("" ::: "memory");  // same-wave DS ops complete in order
    int cnt = min(32, e - base);
    const int2* buf = &ebuf[wslot][0][0];
#pragma unroll 4
    for (int j = 0; j < cnt; ++j) {
      int2 q = buf[j];
      float w = __int_as_float(q.y);
      float2 xv = xin[(size_t)q.x * 32 + lane];
      sum.x = fmaf(w, xv.x, sum.x);
      sum.y = fmaf(w, xv.y, sum.y);
    }
    __builtin_amdgcn_wave_barrier();
    asm volatile("" ::: "memory");
  }
#endif

  size_t o = (size_t)node * 32 + lane;
  xout[o] = sum;
  float2 a = acc[o];
  a.x += sum.x;
  a.y += sum.y;
  acc[o] = a;
}

__global__ void k_final(const float* __restrict__ acc, float* __restrict__ out, int n, float sc) {
  int i = blockIdx.x * blockDim.x + threadIdx.x;
  if (i < n) out[i] = acc[i] * sc;
}

// ---------------- launcher ----------------

extern "C" void kernel_launch(void* const* d_in, const int* in_sizes, int n_in,
                              void* d_out, int out_size, void* d_ws, size_t ws_size,
                              hipStream_t stream) {
  const float* emb = (const float*)d_in[0];   // [N, 64] float32
  const int*   ei  = (const int*)d_in[1];     // [2, E] int32
  const int ND = in_sizes[0];                 // N * 64
  const int N  = ND / 64;
  const int E  = in_sizes[1] / 2;
  const int* row = ei;
  const int* col = ei + E;

  // Carve workspace (256B-aligned regions).
  char*  base = (char*)d_ws;
  size_t cur  = 0;
  auto carve = [&](size_t bytes) -> void* {
    void* p = base + cur;
    cur += (bytes + 255) & ~(size_t)255;
    return p;
  };
  int*   deg       = (int*)  carve((size_t)N * 4);
  float* dinv      = (float*)carve((size_t)N * 4);
  int*   chunkExcl = (int*)  carve((size_t)N * 4);
  int*   part      = (int*)  carve(256 * 4);
  int*   partExcl  = (int*)  carve(256 * 4);
  int*   off       = (int*)  carve((size_t)(N + 1) * 4);
  int*   cursor    = (int*)  carve((size_t)N * 4);
  int2*  pairs     = (int2*) carve((size_t)E * 8);
  float* x0        = (float*)carve((size_t)ND * 4);
  float* x1        = (float*)carve((size_t)ND * 4);
  float* acc       = (float*)carve((size_t)ND * 4);
  (void)ws_size; (void)n_in; (void)out_size;

  const int gN  = (N  + THREADS - 1) / THREADS;
  const int gE  = (E  + THREADS - 1) / THREADS;
  const int gND = (ND + THREADS - 1) / THREADS;
  const int numChunks = (N + 1023) / 1024;

  // --- build CSC + norms (once per launch) ---
  k_zero_i32 <<<gN, THREADS, 0, stream>>>(deg, N);
  k_count_deg<<<gE, THREADS, 0, stream>>>(col, deg, E);
  k_dinv     <<<gN, THREADS, 0, stream>>>(deg, dinv, N);
  k_scan_block<<<numChunks, THREADS, 0, stream>>>(deg, chunkExcl, part, N);
  k_scan_part <<<1, THREADS, 0, stream>>>(part, partExcl, numChunks);
  k_scan_add  <<<gN, THREADS, 0, stream>>>(chunkExcl, partExcl, off, cursor, N, E);
  k_fill      <<<gE, THREADS, 0, stream>>>(row, col, dinv, cursor, pairs, E);
  k_init      <<<gND, THREADS, 0, stream>>>(emb, x0, acc, ND);

  // --- 100 propagation layers, ping-pong x buffers, acc fused ---
  float* xin  = x0;
  float* xout = x1;
  const int gL = (N * 32 + THREADS - 1) / THREADS;  // one wave32 per node
  for (int l = 0; l < LAYERS; ++l) {
    k_layer<<<gL, THREADS, 0, stream>>>((const float2*)xin, (float2*)xout,
                                        (float2*)acc, pairs, off, N);
    float* t = xin; xin = xout; xout = t;
  }

  k_final<<<gND, THREADS, 0, stream>>>(acc, (float*)d_out, ND, 1.0f / (float)(LAYERS + 1));
}